// ModelNew_13211319402746
// MI455X (gfx1250) — compile-verified
//
#include <hip/hip_runtime.h>
#include <stdint.h>

typedef __attribute__((ext_vector_type(2))) float v2f;
typedef __attribute__((ext_vector_type(4))) float v4f;
typedef __attribute__((ext_vector_type(8))) float v8f;

#define NN 4096
#define BM 64
#define BN 64
#define BK 16
#define LDA_S 20            // padded LDS stride for A tile (floats): 80B rows, 16B aligned
#define LDB_S 80            // padded LDS stride for B tile (floats): 320B rows, 16B aligned
#define A_TILE (BM * LDA_S) // 1280 floats
#define B_TILE (BK * LDB_S) // 1280 floats

// ---- CDNA5 async global->LDS copy (ASYNCcnt-tracked, bypasses VGPRs) ----
__device__ __forceinline__ void async_ld_b128(uint32_t lds_addr, const float* g) {
  asm volatile("global_load_async_to_lds_b128 %0, %1, off"
               :: "v"(lds_addr), "v"(g) : "memory");
}
__device__ __forceinline__ void wait_async0() {
  asm volatile("s_wait_asynccnt 0x0" ::: "memory");
}
__device__ __forceinline__ uint32_t lds_addr_of(const float* p) {
  return (uint32_t)(uintptr_t)p;  // generic LDS pointer: low 32 bits = LDS offset
}

// ---- tile fills ----
// masked A fill (needed only when kt == bi): zero A[i,k] where k < i
__device__ __forceinline__ void fill_A_masked(float* As, const float* A,
                                              int rowStart, int kStart, int tid) {
  int f = tid * 4;
#pragma unroll
  for (int it = 0; it < 2; ++it) {
    int ff = f + it * 512;
    int r = ff >> 4, c = ff & 15;
    int gi = rowStart + r, gk = kStart + c;
    v4f a = *(const v4f*)(A + (size_t)gi * NN + gk);
    a.x = (gk + 0 >= gi) ? a.x : 0.f;
    a.y = (gk + 1 >= gi) ? a.y : 0.f;
    a.z = (gk + 2 >= gi) ? a.z : 0.f;
    a.w = (gk + 3 >= gi) ? a.w : 0.f;
    *(v4f*)(As + r * LDA_S + c) = a;
  }
}
// masked B fill (needed only when kt == bj): zero B[k,j] where k > j
__device__ __forceinline__ void fill_B_masked(float* Bs, const float* B,
                                              int colStart, int kStart, int tid) {
  int f = tid * 4;
#pragma unroll
  for (int it = 0; it < 2; ++it) {
    int ff = f + it * 512;
    int r = ff >> 6, c = ff & 63;
    int gk = kStart + r, gj = colStart + c;
    v4f b = *(const v4f*)(B + (size_t)gk * NN + gj);
    b.x = (gk <= gj + 0) ? b.x : 0.f;
    b.y = (gk <= gj + 1) ? b.y : 0.f;
    b.z = (gk <= gj + 2) ? b.z : 0.f;
    b.w = (gk <= gj + 3) ? b.w : 0.f;
    *(v4f*)(Bs + r * LDB_S + c) = b;
  }
}
// interior fills: raw async DMA, no masking needed
__device__ __forceinline__ void fill_A_async(float* As, const float* A,
                                             int rowStart, int kStart, int tid) {
  int f = tid * 4;
#pragma unroll
  for (int it = 0; it < 2; ++it) {
    int ff = f + it * 512;
    int r = ff >> 4, c = ff & 15;
    async_ld_b128(lds_addr_of(As + r * LDA_S + c),
                  A + (size_t)(rowStart + r) * NN + kStart + c);
  }
}
__device__ __forceinline__ void fill_B_async(float* Bs, const float* B,
                                             int colStart, int kStart, int tid) {
  int f = tid * 4;
#pragma unroll
  for (int it = 0; it < 2; ++it) {
    int ff = f + it * 512;
    int r = ff >> 6, c = ff & 63;
    async_ld_b128(lds_addr_of(Bs + r * LDB_S + c),
                  B + (size_t)(kStart + r) * NN + colStart + c);
  }
}

__global__ __launch_bounds__(128) void triu_gemm_wmma_f32(const float* __restrict__ A,
                                                          const float* __restrict__ B,
                                                          float* __restrict__ C) {
  __shared__ float AsBuf[2 * A_TILE];  // double-buffered
  __shared__ float BsBuf[2 * B_TILE];

  const int bi = blockIdx.y;
  const int bj = blockIdx.x;
  const int tid = threadIdx.x;  // 0..127 (4 wave32s)
  const int lane = tid & 31;
  const int wave = tid >> 5;

  const int rowStart = bi * BM;
  const int colStart = bj * BN;

  if (bj < bi) {  // strictly-lower block: exactly zero, but must be written
    float* cbase = C + (size_t)rowStart * NN + colStart;
    const v4f zero = {0.f, 0.f, 0.f, 0.f};
#pragma unroll
    for (int r = 0; r < 8; ++r) {
      int f = (tid + r * 128) * 4;
      *(v4f*)(cbase + (size_t)(f >> 6) * NN + (f & 63)) = zero;
    }
    return;
  }

  v8f acc[2][2] = {};

  const int wm = (wave >> 1) * 32;
  const int wn = (wave & 1) * 32;
  const int halfK = lane >> 4;
  const int l15 = lane & 15;

  // ---- prologue: stage kt = bi into buffer 0 (A edge; B edge iff bi == bj) ----
  fill_A_masked(AsBuf, A, rowStart, bi * BK, tid);
  fill_B_masked(BsBuf, B, colStart, bi * BK, tid);
  __syncthreads();

  for (int kt = bi; kt <= bj; ++kt) {
    const int cur = (kt - bi) & 1;
    const float* Ac = AsBuf + cur * A_TILE;
    const float* Bc = BsBuf + cur * B_TILE;

    // ---- pipelined fill of tile kt+1 into the other buffer ----
    if (kt < bj) {
      const int nxt = cur ^ 1;
      // A is always interior for kt+1 (> bi): pure async DMA
      fill_A_async(AsBuf + nxt * A_TILE, A, rowStart, (kt + 1) * BK, tid);
      if (kt + 1 == bj)
        fill_B_masked(BsBuf + nxt * B_TILE, B, colStart, (kt + 1) * BK, tid);
      else
        fill_B_async(BsBuf + nxt * B_TILE, B, colStart, (kt + 1) * BK, tid);
      if (kt + 2 <= bj) {  // warm L2 one tile deeper (global_prefetch_b8)
        __builtin_prefetch(A + (size_t)(rowStart + (tid & 63)) * NN + (size_t)(kt + 2) * BK, 0, 1);
        __builtin_prefetch(B + ((size_t)(kt + 2) * BK + (tid & 15)) * NN + colStart, 0, 1);
      }
    }

    // ---- compute: 4 k-steps x 2x2 tiles = 16 WMMAs on the current buffer ----
#pragma unroll
    for (int kk = 0; kk < BK; kk += 4) {
      v2f af[2];
#pragma unroll
      for (int tm = 0; tm < 2; ++tm) {
        int row = wm + tm * 16 + l15;
        af[tm] = *(const v2f*)(Ac + row * LDA_S + kk + halfK * 2);  // A 16x4 layout
      }
      v2f bf[2];
#pragma unroll
      for (int tn = 0; tn < 2; ++tn) {
        int col = wn + tn * 16 + l15;
        int k0 = kk + halfK * 2;
        bf[tn].x = Bc[k0 * LDB_S + col];  // B 4x16 layout
        bf[tn].y = Bc[(k0 + 1) * LDB_S + col];
      }
#pragma unroll
      for (int tm = 0; tm < 2; ++tm)
#pragma unroll
        for (int tn = 0; tn < 2; ++tn)
          acc[tm][tn] = __builtin_amdgcn_wmma_f32_16x16x4_f32(
              false, af[tm], false, bf[tn], (short)0, acc[tm][tn], false, false);
    }

    wait_async0();    // next-tile DMA complete (cheap no-op on last iteration)
    __syncthreads();  // one barrier per k-tile (dscnt waits auto-inserted)
  }

  // ---- store with final triu mask (only matters on diagonal blocks) ----
#pragma unroll
  for (int tm = 0; tm < 2; ++tm) {
#pragma unroll
    for (int tn = 0; tn < 2; ++tn) {
      int colg = colStart + wn + tn * 16 + l15;
#pragma unroll
      for (int v = 0; v < 8; ++v) {
        int rowg = rowStart + wm + tm * 16 + v + halfK * 8;
        float val = acc[tm][tn][v];
        C[(size_t)rowg * NN + colg] = (colg >= rowg) ? val : 0.f;
      }
    }
  }
}

extern "C" void kernel_launch(void* const* d_in, const int* in_sizes, int n_in,
                              void* d_out, int out_size, void* d_ws, size_t ws_size,
                              hipStream_t stream) {
  const float* A = (const float*)d_in[0];
  const float* B = (const float*)d_in[1];
  float* C = (float*)d_out;
  dim3 grid(NN / BN, NN / BM);  // x = bj, y = bi
  dim3 block(128);
  triu_gemm_wmma_f32<<<grid, block, 0, stream>>>(A, B, C);
}